// BipartiteGCNStack_712964571492
// MI455X (gfx1250) — compile-verified
//
#include <hip/hip_runtime.h>
#include <hip/hip_bf16.h>

// ---------------------------------------------------------------------------
// BipartiteGCNStack for MI455X (gfx1250, wave32, WGP with 320KB LDS)
//   - fp32 GEMMs via V_WMMA_F32_16X16X4_F32; one wave -> 16x64 output strip
//   - W staged per-block in LDS (32x256 panel, padded stride 272 -> conflict-
//     free ds reads for both lane halves), b128 global loads + ds_store_b128
//   - COO scatter via hardware global_atomic_add_f32 (tables are L2-resident:
//     WH 64MB + msg 64MB << 192MB L2)
//   - BatchNorm batch stats via column-partial sums + fp32 atomics
// ---------------------------------------------------------------------------

typedef float v2f __attribute__((ext_vector_type(2)));
typedef float v8f __attribute__((ext_vector_type(8)));

static constexpr int N_TGT   = 32768;
static constexpr int N_SRC   = 65536;
static constexpr int D       = 256;
static constexpr int N_EDGES = 1048576;
static constexpr float BN_EPS = 1e-5f;

// ---------------------------------------------------------------------------
// GEMM + bias: C[M,256] = A[M,256] @ W[256,256] + bias[256]
// Block = 8 waves covering 2 M-tiles x full N (4 groups of 64).
// Fragment layouts per CDNA5 ISA 7.12.2 (wave32):
//   A 16x4 : lane<16 -> (M=lane, K=kk+0/kk+1), lane>=16 -> (M=lane-16, K=kk+2/kk+3)
//   B 4x16 : VGPR0 = rows kk/kk+2 striped over lanes, VGPR1 = rows kk+1/kk+3
//   C 16x16: VGPR i = row i (lanes 0-15) / row i+8 (lanes 16-31)
// ---------------------------------------------------------------------------
__global__ __launch_bounds__(256) void gemm_bias_f32_wmma(
    const float* __restrict__ A, const float* __restrict__ W,
    const float* __restrict__ bias, float* __restrict__ C, int M)
{
    constexpr int KP  = 32;    // K-panel depth staged in LDS
    constexpr int LDW = 272;   // padded LDS row stride: 2*272 % 64 == 32 ->
                               // lanes 16-31 hit banks 32..47, disjoint from 0..15
    __shared__ __align__(16) float lW[KP * LDW];

    const int tid  = threadIdx.x;
    const int lane = tid & 31;
    const int wave = tid >> 5;
    const int t    = blockIdx.x * 8 + wave;   // global wave-tile id
    const int ng   = t & 3;                   // 64-wide N group
    const int tm   = t >> 2;                  // 16-row M tile (grid is exact)

    const int row     = tm * 16 + (lane & 15);
    const int khalf   = (lane >> 4) * 2;      // 0 for lanes 0-15, 2 for 16-31
    const int colbase = ng * 64 + (lane & 15);

    v8f c0 = {}, c1 = {}, c2 = {}, c3 = {};

    for (int kk = 0; kk < D; kk += KP) {
        if (kk) __syncthreads();              // previous panel fully consumed

        // Stage W[kk..kk+31][0..255]: 8192 floats, 8 float4 per thread.
#pragma unroll
        for (int it = 0; it < (KP * D / 4) / 256; ++it) {
            const int idx  = it * 256 + tid;  // float4 index within panel
            const int grow = idx >> 6;        // 64 float4 per 256-wide row
            const int gc4  = idx & 63;
            const float4 w4 = *(const float4*)(W + (size_t)(kk + grow) * D + gc4 * 4);
            *(float4*)(lW + grow * LDW + gc4 * 4) = w4;
        }
        __syncthreads();

#pragma unroll
        for (int dk = 0; dk < KP; dk += 4) {
            const int k0 = kk + dk + khalf;
            v2f a = *(const v2f*)(A + (size_t)row * D + k0);

            const float* l0 = lW + (dk + khalf) * LDW;
            const float* l1 = l0 + LDW;
            v2f b0 = { l0[colbase +  0], l1[colbase +  0] };
            v2f b1 = { l0[colbase + 16], l1[colbase + 16] };
            v2f b2 = { l0[colbase + 32], l1[colbase + 32] };
            v2f b3 = { l0[colbase + 48], l1[colbase + 48] };

            c0 = __builtin_amdgcn_wmma_f32_16x16x4_f32(false, a, false, b0, (short)0, c0, false, false);
            c1 = __builtin_amdgcn_wmma_f32_16x16x4_f32(false, a, false, b1, (short)0, c1, false, false);
            c2 = __builtin_amdgcn_wmma_f32_16x16x4_f32(false, a, false, b2, (short)0, c2, false, false);
            c3 = __builtin_amdgcn_wmma_f32_16x16x4_f32(false, a, false, b3, (short)0, c3, false, false);
        }
    }

    const int   rowhi = (lane >> 4) * 8;
    const float bb0 = bias[colbase +  0];
    const float bb1 = bias[colbase + 16];
    const float bb2 = bias[colbase + 32];
    const float bb3 = bias[colbase + 48];

#pragma unroll
    for (int i = 0; i < 8; ++i) {
        const int r = tm * 16 + rowhi + i;
        float* Crow = C + (size_t)r * D;
        Crow[colbase +  0] = c0[i] + bb0;
        Crow[colbase + 16] = c1[i] + bb1;
        Crow[colbase + 32] = c2[i] + bb2;
        Crow[colbase + 48] = c3[i] + bb3;
    }
}

// ---------------------------------------------------------------------------
// deg[dst[e]] += vals[e]
// ---------------------------------------------------------------------------
__global__ __launch_bounds__(256) void deg_edges(
    const float* __restrict__ vals, const int* __restrict__ dst,
    float* __restrict__ deg, int n_edges)
{
    int e = blockIdx.x * 256 + threadIdx.x;
    if (e < n_edges) unsafeAtomicAdd(deg + dst[e], vals[e]);
}

// ---------------------------------------------------------------------------
// msg[dst[e], :] += vals[e] * WH[src[e], :]   (one wave per edge, 8 f32/lane)
// ---------------------------------------------------------------------------
__global__ __launch_bounds__(256) void scatter_edges(
    const float* __restrict__ WH, const float* __restrict__ vals,
    const int* __restrict__ src, const int* __restrict__ dst,
    float* __restrict__ msg, int n_edges)
{
    const int e = blockIdx.x * 8 + (threadIdx.x >> 5);
    if (e >= n_edges) return;
    const int lane = threadIdx.x & 31;
    const int s = src[e];
    const int d = dst[e];
    const float v = vals[e];
    const float* in  = WH  + (size_t)s * D;
    float*       out = msg + (size_t)d * D;
#pragma unroll
    for (int i = lane; i < D; i += 32)
        unsafeAtomicAdd(out + i, v * in[i]);
}

// ---------------------------------------------------------------------------
// x = relu(msg / max(deg,1)) + H_tgt ; accumulate per-column sum/sumsq.
// One block per 128-row strip, thread == column (coalesced rows).
// ---------------------------------------------------------------------------
__global__ __launch_bounds__(256) void bn_stats(
    const float* __restrict__ msg, const float* __restrict__ deg,
    const float* __restrict__ Htgt, float* __restrict__ x,
    float* __restrict__ stats, int rows_per_block)
{
    const int col = threadIdx.x;
    const int r0  = blockIdx.x * rows_per_block;
    float s = 0.f, s2 = 0.f;
    for (int r = r0; r < r0 + rows_per_block; ++r) {
        float dg = deg[r];
        dg = dg > 1.f ? dg : 1.f;
        float m = msg[(size_t)r * D + col] / dg;
        m = m > 0.f ? m : 0.f;
        const float xv = m + Htgt[(size_t)r * D + col];
        x[(size_t)r * D + col] = xv;
        s  += xv;
        s2 += xv * xv;
    }
    unsafeAtomicAdd(stats + col,     s);
    unsafeAtomicAdd(stats + D + col, s2);
}

// ---------------------------------------------------------------------------
// Htgt = gamma * (x - mu) / sqrt(var + eps) + beta
// ---------------------------------------------------------------------------
__global__ __launch_bounds__(256) void bn_apply(
    const float* __restrict__ x, const float* __restrict__ stats,
    const float* __restrict__ gamma, const float* __restrict__ beta,
    float* __restrict__ Htgt)
{
    const int idx = blockIdx.x * 256 + threadIdx.x;
    const int col = idx & (D - 1);
    const float invN = 1.f / (float)N_TGT;
    const float mu  = stats[col] * invN;
    const float var = stats[D + col] * invN - mu * mu;
    const float inv = rsqrtf(var + BN_EPS);
    Htgt[idx] = gamma[col] * (x[idx] - mu) * inv + beta[col];
}

// ---------------------------------------------------------------------------
// out = relu(msg / max(deg,1))  (backward direction result, N_SRC rows)
// ---------------------------------------------------------------------------
__global__ __launch_bounds__(256) void relu_div(
    const float* __restrict__ msg, const float* __restrict__ deg,
    float* __restrict__ out)
{
    const int idx = blockIdx.x * 256 + threadIdx.x;
    float dg = deg[idx >> 8];
    dg = dg > 1.f ? dg : 1.f;
    float m = msg[idx] / dg;
    out[idx] = m > 0.f ? m : 0.f;
}

// ---------------------------------------------------------------------------
extern "C" void kernel_launch(void* const* d_in, const int* in_sizes, int n_in,
                              void* d_out, int out_size, void* d_ws, size_t ws_size,
                              hipStream_t stream)
{
    (void)in_sizes; (void)n_in; (void)out_size; (void)ws_size;

    const float* H_src_in  = (const float*)d_in[0];
    const float* target_in = (const float*)d_in[1];
    const float* W_fwd     = (const float*)d_in[2];
    const float* b_fwd     = (const float*)d_in[3];
    const float* W_bwd     = (const float*)d_in[4];
    const float* b_bwd     = (const float*)d_in[5];
    const float* gamma     = (const float*)d_in[6];
    const float* beta      = (const float*)d_in[7];
    const float* vals      = (const float*)d_in[8];
    const int*   rows      = (const int*)d_in[9];
    const int*   cols      = (const int*)d_in[10];
    float*       Htgt      = (float*)d_out;       // live H_tgt, also final output

    // Workspace layout (floats). x-buffer aliases WH (WH dead by BN time).
    float* ws   = (float*)d_ws;
    float* Hsrc = ws;                               // N_SRC * D
    float* WH   = Hsrc + (size_t)N_SRC * D;         // N_SRC * D  (alias: BN x)
    float* MSG  = WH   + (size_t)N_SRC * D;         // N_SRC * D
    float* DEG  = MSG  + (size_t)N_SRC * D;         // N_SRC
    float* STAT = DEG  + N_SRC;                     // 2 * D

    hipMemcpyAsync(Hsrc, H_src_in, (size_t)N_SRC * D * sizeof(float),
                   hipMemcpyDeviceToDevice, stream);
    hipMemcpyAsync(Htgt, target_in, (size_t)N_TGT * D * sizeof(float),
                   hipMemcpyDeviceToDevice, stream);

    const int gemm_blocks_src = (N_SRC / 16) * 4 / 8;   // 2048
    const int gemm_blocks_tgt = (N_TGT / 16) * 4 / 8;   // 1024

    for (int l = 0; l < 2; ++l) {
        const float* Wf = W_fwd + (size_t)l * D * D;
        const float* bf = b_fwd + (size_t)l * D;
        const float* gl = gamma + (size_t)l * D;
        const float* bl = beta  + (size_t)l * D;

        // ---- forward: targets aggregate from sources -----------------------
        gemm_bias_f32_wmma<<<gemm_blocks_src, 256, 0, stream>>>(Hsrc, Wf, bf, WH, N_SRC);

        hipMemsetAsync(MSG, 0, (size_t)N_TGT * D * sizeof(float), stream);
        hipMemsetAsync(DEG, 0, (size_t)N_TGT * sizeof(float), stream);
        deg_edges<<<N_EDGES / 256, 256, 0, stream>>>(vals, rows, DEG, N_EDGES);
        scatter_edges<<<N_EDGES / 8, 256, 0, stream>>>(WH, vals, cols, rows, MSG, N_EDGES);

        // ---- residual + BatchNorm -----------------------------------------
        hipMemsetAsync(STAT, 0, 2 * D * sizeof(float), stream);
        bn_stats<<<256, 256, 0, stream>>>(MSG, DEG, Htgt, /*x=*/WH, STAT, N_TGT / 256);
        bn_apply<<<(size_t)N_TGT * D / 256, 256, 0, stream>>>(WH, STAT, gl, bl, Htgt);

        // ---- backward: sources aggregate from targets (dead for last layer)
        if (l == 0) {
            const float* Wb = W_bwd + (size_t)l * D * D;
            const float* bb = b_bwd + (size_t)l * D;

            gemm_bias_f32_wmma<<<gemm_blocks_tgt, 256, 0, stream>>>(Htgt, Wb, bb, WH, N_TGT);

            hipMemsetAsync(MSG, 0, (size_t)N_SRC * D * sizeof(float), stream);
            hipMemsetAsync(DEG, 0, (size_t)N_SRC * sizeof(float), stream);
            deg_edges<<<N_EDGES / 256, 256, 0, stream>>>(vals, cols, DEG, N_EDGES);
            scatter_edges<<<N_EDGES / 8, 256, 0, stream>>>(WH, vals, rows, cols, MSG, N_EDGES);
            relu_div<<<(size_t)N_SRC * D / 256, 256, 0, stream>>>(MSG, DEG, Hsrc);
        }
    }
}